// CLUBLoss_27986006901387
// MI455X (gfx1250) — compile-verified
//
#include <hip/hip_runtime.h>
#include <hip/hip_bf16.h>

typedef __attribute__((ext_vector_type(2))) float v2f;
typedef __attribute__((ext_vector_type(8))) float v8f;

#define NROWS 1024
#define DDIM  256
#define HDIM  512

// ---------------------------------------------------------------------------
// GEMM: out[M x Ncols] = act(A[M x K] @ W[K x Ncols] + bias[Ncols])
// One wave32 computes one 16x16 tile via V_WMMA_F32_16X16X4_F32.
// Grid must exactly cover (M/16)*(Ncols/16) tiles with blockDim.x/32 waves/blk.
// ---------------------------------------------------------------------------
__global__ __launch_bounds__(256) void gemm_wmma_f32_kernel(
    const float* __restrict__ A, const float* __restrict__ W,
    const float* __restrict__ bias, float* __restrict__ out,
    int M, int K, int Ncols, int relu)
{
    const int lane   = threadIdx.x & 31;
    const int wave   = blockIdx.x * (blockDim.x >> 5) + (threadIdx.x >> 5);
    const int tilesN = Ncols >> 4;
    const int tm     = wave / tilesN;
    const int tn     = wave - tm * tilesN;

    const int rowA = tm * 16 + (lane & 15);     // A row for this lane
    const int col  = tn * 16 + (lane & 15);     // B/C column for this lane
    const int kgrp = (lane >> 4) << 1;          // lanes 16-31 handle K+2,K+3

    const float* __restrict__ Arow = A + (size_t)rowA * K;

    v8f acc = {0.f, 0.f, 0.f, 0.f, 0.f, 0.f, 0.f, 0.f};

#pragma unroll 4
    for (int k0 = 0; k0 < K; k0 += 4) {
        const int kb = k0 + kgrp;
        // A fragment: two contiguous fp32 along K (8B-aligned: kb is even)
        v2f a = *reinterpret_cast<const v2f*>(Arow + kb);
        // B fragment: W[kb][col], W[kb+1][col] (coalesced across lanes)
        v2f b;
        b.x = W[(size_t)kb * Ncols + col];
        b.y = W[(size_t)(kb + 1) * Ncols + col];
        acc = __builtin_amdgcn_wmma_f32_16x16x4_f32(
            /*neg_a=*/false, a, /*neg_b=*/false, b,
            /*c_mod=*/(short)0, acc, /*reuse_a=*/false, /*reuse_b=*/false);
    }

    const float bv    = bias[col];
    const int   rbase = tm * 16 + ((lane >> 4) << 3);
#pragma unroll
    for (int r = 0; r < 8; ++r) {
        float v = acc[r] + bv;
        if (relu) v = fmaxf(v, 0.0f);
        out[(size_t)(rbase + r) * Ncols + col] = v;
    }
}

// ---------------------------------------------------------------------------
// Column stats of domain_b: mb[d] = mean_j b[j,d], mb2[d] = mean_j b[j,d]^2
// ---------------------------------------------------------------------------
__global__ __launch_bounds__(256) void bstats_kernel(
    const float* __restrict__ B, float* __restrict__ mb,
    float* __restrict__ mb2, int N, int D)
{
    int d = blockIdx.x * blockDim.x + threadIdx.x;
    if (d >= D) return;
    float s = 0.f, s2 = 0.f;
    for (int j = 0; j < N; ++j) {
        float v = B[(size_t)j * D + d];   // coalesced across d
        s  += v;
        s2 += v * v;
    }
    float inv = 1.0f / (float)N;
    mb[d]  = s * inv;
    mb2[d] = s2 * inv;
}

// ---------------------------------------------------------------------------
// Per-row epilogue: l2norm(mu), tanh(lv), pos/neg/prob terms -> row_part[i]
// One block (256 threads == D) per row.
// ---------------------------------------------------------------------------
__global__ __launch_bounds__(256) void finalize_kernel(
    const float* __restrict__ o_mu, const float* __restrict__ o_lv,
    const float* __restrict__ B, const float* __restrict__ mb,
    const float* __restrict__ mb2, float* __restrict__ row_part, int D)
{
    __shared__ float sdata[256];
    const int i = blockIdx.x;
    const int d = threadIdx.x;

    const float m = o_mu[(size_t)i * D + d];
    sdata[d] = m * m;
    __syncthreads();
#pragma unroll
    for (int s = 128; s > 0; s >>= 1) {
        if (d < s) sdata[d] += sdata[d + s];
        __syncthreads();
    }
    const float nrm = fmaxf(sqrtf(sdata[0]), 1e-12f);
    __syncthreads();   // everyone has read sdata[0] before it is reused

    const float mu   = m / nrm;
    const float lv   = tanhf(o_lv[(size_t)i * D + d]);
    const float iv   = expf(-lv);
    const float bi   = B[(size_t)i * D + d];
    const float diff = bi - mu;

    const float prob = diff * diff * iv + lv;        // prob-model term
    const float pos  = -prob;                        // pos_probs term
    const float d2m  = mb2[d] - 2.f * mu * mb[d] + mu * mu; // mean_j (b_j - mu)^2
    const float neg  = -d2m * iv - lv;               // neg_probs term

    sdata[d] = (pos - neg) + prob;                   // embed + prob contribution
    __syncthreads();
#pragma unroll
    for (int s = 128; s > 0; s >>= 1) {
        if (d < s) sdata[d] += sdata[d + s];
        __syncthreads();
    }
    if (d == 0) row_part[i] = sdata[0];
}

// ---------------------------------------------------------------------------
// Deterministic final reduction: loss = mean_i row_part[i]
// ---------------------------------------------------------------------------
__global__ __launch_bounds__(256) void reduce_kernel(
    const float* __restrict__ row_part, float* __restrict__ out, int N)
{
    __shared__ float sdata[256];
    const int t = threadIdx.x;
    float s = 0.f;
    for (int i = t; i < N; i += 256) s += row_part[i];
    sdata[t] = s;
    __syncthreads();
#pragma unroll
    for (int k = 128; k > 0; k >>= 1) {
        if (t < k) sdata[t] += sdata[t + k];
        __syncthreads();
    }
    if (t == 0) out[0] = sdata[0] / (float)N;
}

// ---------------------------------------------------------------------------
extern "C" void kernel_launch(void* const* d_in, const int* in_sizes, int n_in,
                              void* d_out, int out_size, void* d_ws, size_t ws_size,
                              hipStream_t stream) {
    (void)in_sizes; (void)n_in; (void)out_size; (void)ws_size;

    const float* domain_a = (const float*)d_in[0];
    const float* domain_b = (const float*)d_in[1];
    const float* mu_w0 = (const float*)d_in[2];
    const float* mu_b0 = (const float*)d_in[3];
    const float* mu_w1 = (const float*)d_in[4];
    const float* mu_b1 = (const float*)d_in[5];
    const float* mu_w2 = (const float*)d_in[6];
    const float* mu_b2 = (const float*)d_in[7];
    const float* lv_w0 = (const float*)d_in[8];
    const float* lv_b0 = (const float*)d_in[9];
    const float* lv_w1 = (const float*)d_in[10];
    const float* lv_b1 = (const float*)d_in[11];
    const float* lv_w2 = (const float*)d_in[12];
    const float* lv_b2 = (const float*)d_in[13];

    // Workspace layout (floats): ~1.58M floats (~6.3 MB)
    float* ws       = (float*)d_ws;
    float* hA       = ws;                          // 1024*512
    float* hB       = hA + NROWS * HDIM;           // 1024*512
    float* o_mu     = hB + NROWS * HDIM;           // 1024*256
    float* o_lv     = o_mu + NROWS * DDIM;         // 1024*256
    float* mb       = o_lv + NROWS * DDIM;         // 256
    float* mb2      = mb + DDIM;                   // 256
    float* row_part = mb2 + DDIM;                  // 1024

    // column stats of domain_b
    bstats_kernel<<<1, 256, 0, stream>>>(domain_b, mb, mb2, NROWS, DDIM);

    // grid sizes: tiles/(8 waves per 256-thread block)
    const int blkL01 = (NROWS / 16) * (HDIM / 16) / 8;  // 256 blocks
    const int blkL2  = (NROWS / 16) * (DDIM / 16) / 8;  // 128 blocks

    // mu net
    gemm_wmma_f32_kernel<<<blkL01, 256, 0, stream>>>(domain_a, mu_w0, mu_b0, hA,
                                                     NROWS, DDIM, HDIM, 1);
    gemm_wmma_f32_kernel<<<blkL01, 256, 0, stream>>>(hA, mu_w1, mu_b1, hB,
                                                     NROWS, HDIM, HDIM, 1);
    gemm_wmma_f32_kernel<<<blkL2, 256, 0, stream>>>(hB, mu_w2, mu_b2, o_mu,
                                                    NROWS, HDIM, DDIM, 0);
    // lv net (reuse hA/hB)
    gemm_wmma_f32_kernel<<<blkL01, 256, 0, stream>>>(domain_a, lv_w0, lv_b0, hA,
                                                     NROWS, DDIM, HDIM, 1);
    gemm_wmma_f32_kernel<<<blkL01, 256, 0, stream>>>(hA, lv_w1, lv_b1, hB,
                                                     NROWS, HDIM, HDIM, 1);
    gemm_wmma_f32_kernel<<<blkL2, 256, 0, stream>>>(hB, lv_w2, lv_b2, o_lv,
                                                    NROWS, HDIM, DDIM, 0);

    // epilogue + deterministic reduction
    finalize_kernel<<<NROWS, 256, 0, stream>>>(o_mu, o_lv, domain_b, mb, mb2,
                                               row_part, DDIM);
    reduce_kernel<<<1, 256, 0, stream>>>(row_part, (float*)d_out, NROWS);
}